// Bfunction_69312182222975
// MI455X (gfx1250) — compile-verified
//
#include <hip/hip_runtime.h>

typedef __attribute__((ext_vector_type(2))) float v2f;
typedef __attribute__((ext_vector_type(8))) float v8f;

#define B_DIM 1024
#define M_DIM 256
#define L_DIM 256

// ---------------------------------------------------------------------------
// Precompute 0.5*cos(x), 0.5*sin(x) per (b,m), stored m-major for coalesced
// A-operand loads: tx[m*1024 + b].
// ---------------------------------------------------------------------------
__global__ __launch_bounds__(256) void trig_x_kernel(const float* __restrict__ x,
                                                     v2f* __restrict__ tx) {
    int i = blockIdx.x * 256 + threadIdx.x;   // 0 .. 262143
    int m = i >> 10;                          // / 1024
    int b = i & 1023;
    float xv = x[b * M_DIM + m];
    float s, c;
    __sincosf(xv, &s, &c);
    v2f r = {0.5f * c, 0.5f * s};
    tx[i] = r;
}

// cos(p), sin(p) per (m,l): tp[m*256 + l] (same layout as phis, fully coalesced)
__global__ __launch_bounds__(256) void trig_p_kernel(const float* __restrict__ phis,
                                                     v2f* __restrict__ tp) {
    int i = blockIdx.x * 256 + threadIdx.x;   // 0 .. 65535
    float pv = phis[i];
    float s, c;
    __sincosf(pv, &s, &c);
    v2f r = {c, s};
    tp[i] = r;
}

// ---------------------------------------------------------------------------
// Main kernel: per m, out[b, l] = mask * (0.5*cos(x_b - p_l) + 0.5)
// computed as rank-2 GEMM via V_WMMA_F32_16X16X4_F32 (K=0,1 used; K=2,3 zero).
// Each wave: one 16-row b-tile, loops over all 16 l-tiles (A reused in regs).
// ---------------------------------------------------------------------------
__global__ __launch_bounds__(256) void cos_mask_wmma_kernel(
    const float* __restrict__ x,
    const float* __restrict__ phis,
    const float* __restrict__ interval,
    const v2f* __restrict__ tx,
    const v2f* __restrict__ tp,
    float* __restrict__ out) {

    const int m    = blockIdx.y;              // 0..255
    const int wave = threadIdx.x >> 5;        // 0..7
    const int lane = threadIdx.x & 31;        // wave32
    const int btile = blockIdx.x * 8 + wave;  // 0..63
    const int b0    = btile * 16;
    const bool lo   = (lane < 16);
    const int  ln   = lane & 15;

    // A operand: lanes 0-15 hold (0.5cos x, 0.5sin x) for rows b0+lane (K=0,1);
    // lanes 16-31 are K=2,3 -> zero.
    v2f a = {0.0f, 0.0f};
    float xval = 0.0f;
    if (lo) {
        a    = tx[m * B_DIM + b0 + lane];
        xval = x[(b0 + lane) * M_DIM + m];
    }

    // Broadcast the 16 raw x values across the wave for the mask:
    // lanes 0-15 need rows 0..7, lanes 16-31 need rows 8..15.
    float xr[8];
    const int base = lo ? 0 : 8;
#pragma unroll
    for (int j = 0; j < 8; ++j) {
        xr[j] = __int_as_float(
            __builtin_amdgcn_ds_bpermute((base + j) * 4, __float_as_int(xval)));
    }

    const float iv  = interval[m];
    const float niv = -iv;
    const size_t mco = (size_t)m * L_DIM;

#pragma unroll 1
    for (int lt = 0; lt < 16; ++lt) {
        const int l0 = lt * 16;

        // B operand: lanes 0-15 hold (cos p, sin p) for cols l0+lane (K=0,1);
        // lanes 16-31 zero.
        v2f bm = {0.0f, 0.0f};
        if (lo) bm = tp[m * L_DIM + l0 + lane];

        // Raw phi for the mask: column = lane&15 in both lane halves.
        const float phi = phis[m * L_DIM + l0 + ln];

        v8f c = {};
        // D = A x B + 0 : rank-2 update -> 0.5*cos(x_b - p_l)
        c = __builtin_amdgcn_wmma_f32_16x16x4_f32(
            /*neg_a=*/false, a, /*neg_b=*/false, bm,
            /*c_mod=*/(short)0, c, /*reuse_a=*/false, /*reuse_b=*/false);

#pragma unroll
        for (int j = 0; j < 8; ++j) {
            const float diff = xr[j] - phi;
            const float val  = c[j] + 0.5f;
            const float r    = (diff > niv && diff <= iv) ? val : 0.0f;
            const int brow   = b0 + base + j;
            out[(size_t)brow * (M_DIM * L_DIM) + mco + l0 + ln] = r;
        }
    }
}

extern "C" void kernel_launch(void* const* d_in, const int* in_sizes, int n_in,
                              void* d_out, int out_size, void* d_ws, size_t ws_size,
                              hipStream_t stream) {
    const float* x        = (const float*)d_in[0];
    const float* phis     = (const float*)d_in[1];
    const float* interval = (const float*)d_in[2];
    float* out            = (float*)d_out;

    // Workspace layout: tx = 262144 v2f (2 MB) @ 0, tp = 65536 v2f (512 KB) @ 2 MB
    v2f* tx = (v2f*)d_ws;
    v2f* tp = (v2f*)((char*)d_ws + (size_t)B_DIM * M_DIM * sizeof(v2f));

    trig_x_kernel<<<dim3((B_DIM * M_DIM) / 256), dim3(256), 0, stream>>>(x, tx);
    trig_p_kernel<<<dim3((M_DIM * L_DIM) / 256), dim3(256), 0, stream>>>(phis, tp);

    // grid: x = 8 blocks of 8 waves -> 64 b-tiles; y = 256 (m)
    cos_mask_wmma_kernel<<<dim3(8, M_DIM), dim3(256), 0, stream>>>(
        x, phis, interval, tx, tp, out);
}